// Conv_FastLIF_FastLIF_69277822485156
// MI455X (gfx1250) — compile-verified
//
#include <hip/hip_runtime.h>

// Conv(5x5, SAME) + FastLIF + FastLI fused scan for MI455X (gfx1250).
// Memory-bound (512 MB min traffic @ 23.3 TB/s ~= 22us). One wave per 16x16
// pixel tile; LIF states live in registers across all 256 timesteps; per-t
// 20x20 halo tile is staged global->LDS with double-buffered async loads;
// conv is done as 25 x V_WMMA_F32_16X16X4_F32 (exact f32, banded-B formulation).

typedef __attribute__((ext_vector_type(2))) float v2f;
typedef __attribute__((ext_vector_type(8))) float v8f;

#define T_STEPS 256
#define HH 512
#define WW 512
#define HWPIX (HH * WW)
#define TILE 16
#define HALO 20            // TILE + 4
#define TILE_ELEMS 400     // 20*20
#define N_ISSUE 13         // ceil(400/32)

#define ALPHA_LIF 0.85f
#define V_TH 2.0f
#define ALPHA_LI 0.9f

__device__ __forceinline__ void async_load_b32(unsigned long long base, int goff,
                                               unsigned ldsaddr) {
  // GVS mode: mem = SGPR64 base + signed VGPR32 offset; VDST holds LDS byte addr.
  asm volatile("global_load_async_to_lds_b32 %0, %1, %2"
               :: "v"(ldsaddr), "v"(goff), "s"(base)
               : "memory");
}

__global__ __launch_bounds__(32)
void snn_conv_lif_kernel(const float* __restrict__ x,
                         const float* __restrict__ kw,
                         float* __restrict__ out) {
  __shared__ float lds_f[2 * TILE_ELEMS + 4];  // two tile buffers + dummy slot

  const int lane = (int)threadIdx.x;  // 0..31 (wave32)
  const int n  = lane & 15;           // column (N) / row (M) for A-frag
  const int kh = lane >> 4;           // lane half selects K sub-pair
  const int x0 = (int)blockIdx.x * TILE;
  const int y0 = (int)blockIdx.y * TILE;

  // Zero LDS once (covers halo padding + dummy slot); padding positions are
  // never overwritten because out-of-image lanes redirect to the dummy slot.
  for (int i = lane; i < 2 * TILE_ELEMS + 4; i += 32) lds_f[i] = 0.0f;
  asm volatile("s_wait_dscnt 0" ::: "memory");  // DS stores vs async LDS writes

  const unsigned lds_base   = (unsigned)(unsigned long long)(&lds_f[0]);
  const unsigned dummy_addr = lds_base + (unsigned)(2 * TILE_ELEMS) * 4u;

  // Precompute banded B fragments: B_dy[c][x] = k[dy][c-x] for 0<=c-x<=4.
  // B 4x16 layout: VGPR0 = K rows {2*kh}, VGPR1 = K rows {2*kh+1}, N = lane&15.
  v2f bf[5][5];
#pragma unroll
  for (int dy = 0; dy < 5; ++dy) {
#pragma unroll
    for (int j = 0; j < 5; ++j) {
      const int c0 = 4 * j + 2 * kh;
      const int d0 = c0 - n;
      const int d1 = d0 + 1;
      const float bx = (d0 >= 0 && d0 <= 4) ? kw[dy * 5 + d0] : 0.0f;
      const float by = (d1 >= 0 && d1 <= 4) ? kw[dy * 5 + d1] : 0.0f;
      v2f b; b.x = bx; b.y = by;
      bf[dy][j] = b;
    }
  }

  // Issue the 13 async loads that fill one 20x20 halo tile for timestep t.
  auto issue = [&](int t, int buf) {
    const unsigned long long base =
        (unsigned long long)(size_t)x + (unsigned long long)t * (unsigned long long)(HWPIX * 4);
    const unsigned bufaddr = lds_base + (unsigned)buf * (unsigned)(TILE_ELEMS * 4);
#pragma unroll
    for (int it = 0; it < N_ISSUE; ++it) {
      const int e = it * 32 + lane;
      const int r = e / 20;
      const int c = e - r * 20;
      const int gy = y0 - 2 + r;
      const int gx = x0 - 2 + c;
      const bool ok = (e < TILE_ELEMS) & (gy >= 0) & (gy < HH) & (gx >= 0) & (gx < WW);
      const int goff = ok ? ((gy << 9) + gx) * 4 : 0;                    // in-range fallback
      const unsigned laddr = ok ? (bufaddr + (unsigned)(r * HALO + c) * 4u)
                                : dummy_addr;                            // absorb junk
      async_load_b32(base, goff, laddr);  // full EXEC -> always 13 outstanding
    }
  };

  issue(0, 0);

  v8f s1 = {0.f, 0.f, 0.f, 0.f, 0.f, 0.f, 0.f, 0.f};
  v8f s2 = {0.f, 0.f, 0.f, 0.f, 0.f, 0.f, 0.f, 0.f};
  const int m = lane & 15;  // A-frag row

  for (int t = 0; t < T_STEPS; ++t) {
    const int cur = t & 1;
    if (t + 1 < T_STEPS) {
      issue(t + 1, cur ^ 1);
      // 26 outstanding -> wait to 13: current buffer fully landed (in-order done).
      asm volatile("s_wait_asynccnt 13" ::: "memory");
    } else {
      asm volatile("s_wait_asynccnt 0" ::: "memory");
    }

    const float* tile = &lds_f[cur * TILE_ELEMS];

    // Out(16x16) = sum_{dy=0..4} A_dy(16x20) x B_dy(20x16), K split into 5x K=4.
    // A 16x4 f32 layout: lanes 0-15 K={0,1} in v0,v1; lanes 16-31 K={2,3}.
    v8f acc = {0.f, 0.f, 0.f, 0.f, 0.f, 0.f, 0.f, 0.f};
#pragma unroll
    for (int dy = 0; dy < 5; ++dy) {
#pragma unroll
      for (int j = 0; j < 5; ++j) {
        const v2f a = *(const v2f*)(tile + (m + dy) * HALO + 4 * j + 2 * kh);
        acc = __builtin_amdgcn_wmma_f32_16x16x4_f32(
            /*neg_a=*/false, a, /*neg_b=*/false, bf[dy][j],
            /*c_mod=*/(short)0, acc, /*reuse_a=*/false, /*reuse_b=*/false);
      }
    }

    // LIF fire + soft reset, LI readout; store s2 (C/D layout: vgpr i -> row
    // y0 + i + 8*kh, lane -> col x0 + n).
    const size_t obase =
        ((size_t)t * HH + (size_t)(y0 + 8 * kh)) * WW + (size_t)(x0 + n);
#pragma unroll
    for (int i = 0; i < 8; ++i) {
      const float v = ALPHA_LIF * s1[i] + acc[i];
      const float spk = (v >= V_TH) ? 1.0f : 0.0f;
      s1[i] = v - spk * V_TH;
      const float s2n = ALPHA_LI * s2[i] + spk;
      s2[i] = s2n;
      out[obase + (size_t)i * WW] = s2n;
    }
  }
}

extern "C" void kernel_launch(void* const* d_in, const int* in_sizes, int n_in,
                              void* d_out, int out_size, void* d_ws, size_t ws_size,
                              hipStream_t stream) {
  (void)in_sizes; (void)n_in; (void)d_ws; (void)ws_size; (void)out_size;
  const float* x  = (const float*)d_in[0];   // [256,1,512,512] f32
  const float* kw = (const float*)d_in[1];   // [1,1,5,5] f32
  float* out = (float*)d_out;                // [256,1,512,512] f32

  dim3 grid(WW / TILE, HH / TILE, 1);        // 32 x 32 tiles, one wave each
  snn_conv_lif_kernel<<<grid, 32, 0, stream>>>(x, kw, out);
}